// CoMPILE_36249523978270
// MI455X (gfx1250) — compile-verified
//
#include <hip/hip_runtime.h>

// -----------------------------------------------------------------------------
// CustomRGCNConv (mean aggr) for MI455X / gfx1250, wave32.
//   r_weight = edge_emb @ l_weight        -> V_WMMA_F32_16X16X4_F32 (f32-exact)
//   mout     = r_weight * x[src]          -> coalesced gather (x fits in L2)
//   msg_sum  = scatter-add over dst       -> native global_atomic_add_f32
//   out      = msg_sum/max(deg,1) + x@root + bias  -> second WMMA GEMM
// -----------------------------------------------------------------------------

typedef float v2f __attribute__((ext_vector_type(2)));
typedef float v8f __attribute__((ext_vector_type(8)));

#define DIM 64
#define LDS_PITCH 65   // 64+1 padding -> conflict-free column access

// D = A(16x4) * B(4x16) + C, f32.
__device__ __forceinline__ v8f wmma_f32_16x16x4(v2f a, v2f b, v8f c) {
  return __builtin_amdgcn_wmma_f32_16x16x4_f32(
      /*neg_a=*/false, a, /*neg_b=*/false, b,
      /*c_mod=*/(short)0, c, /*reuse_a=*/false, /*reuse_b=*/false);
}

// Native no-return f32 atomic add (GLOBAL_ATOMIC_ADD_F32, STOREcnt-tracked).
// Workspace is ordinary coarse-grained device memory, so unsafeAtomicAdd is
// exact here and avoids the CAS-loop expansion of generic fetch_add.
__device__ __forceinline__ void atomic_add_f32(float* p, float v) {
  unsafeAtomicAdd(p, v);
}

__global__ void zero_ws_kernel(float* __restrict__ p, long long n) {
  long long i = (long long)blockIdx.x * blockDim.x + threadIdx.x;
  if (i < n) p[i] = 0.0f;
}

// ---------------------------------------------------------------------------
// Edge kernel: one wave per 16-edge tile; 8 waves (256 thr) per block.
// ---------------------------------------------------------------------------
__global__ __launch_bounds__(256) void rgcn_edge_kernel(
    const float* __restrict__ x,      // [N,64]
    const int*   __restrict__ ei,     // [2,E] flat: src then dst
    const float* __restrict__ emb,    // [E,64]
    const float* __restrict__ lw,     // [64,64]
    float* __restrict__ msg,          // [N,64] scatter target
    float* __restrict__ deg,          // [N]
    int n_edges)
{
  __shared__ float slwT[DIM * LDS_PITCH];   // lw transposed: slwT[c*65+r]=lw[r][c]
  for (int idx = threadIdx.x; idx < DIM * DIM; idx += 256) {
    int r = idx >> 6, c = idx & 63;
    slwT[c * LDS_PITCH + r] = lw[idx];
  }
  __syncthreads();

  const int lane = threadIdx.x & 31;
  const int lh   = lane & 15;   // lane within half-wave
  const int hi   = lane >> 4;   // half select
  const long long tile  = (long long)blockIdx.x * 8 + (threadIdx.x >> 5);
  const long long ebase = tile * 16;
  if (ebase >= n_edges) return;                 // wave-uniform: EXEC stays all-1s

  // A fragment source: row M = lh of the tile, K pair offset 2*hi (ISA layout:
  // lanes0-15 -> {K0,K1}, lanes16-31 -> {K2,K3} per VGPR pair).
  long long arow_e = ebase + lh;
  if (arow_e >= n_edges) arow_e = n_edges - 1;  // safe clamp (E % 16 == 0 anyway)
  const float* arow = emb + arow_e * DIM + 2 * hi;

  v8f acc[4] = {v8f{}, v8f{}, v8f{}, v8f{}};

#pragma unroll
  for (int k = 0; k < DIM; k += 4) {
    v2f a;
    a.x = arow[k];                    // K = k + 2*hi
    a.y = arow[k + 1];                // K = k + 2*hi + 1
    const int kk = k + 2 * hi;
#pragma unroll
    for (int n = 0; n < 4; ++n) {     // 4 column tiles of 16 -> DIM=64
      v2f b;                          // B: lane = N, VGPR0/1 = K pair (mirror of A)
      b.x = slwT[(n * 16 + lh) * LDS_PITCH + kk];
      b.y = slwT[(n * 16 + lh) * LDS_PITCH + kk + 1];
      acc[n] = wmma_f32_16x16x4(a, b, acc[n]);
    }
  }

  // C layout: VGPR j -> M = j + 8*hi, N = 16n + lh. Pre-load this lane's edges.
  int s_idx[8], t_idx[8];
#pragma unroll
  for (int j = 0; j < 8; ++j) {
    long long e = ebase + j + 8 * hi;
    long long ec = (e < n_edges) ? e : (n_edges - 1);
    s_idx[j] = ei[ec];                       // src
    t_idx[j] = ei[(long long)n_edges + ec];  // dst
  }

#pragma unroll
  for (int n = 0; n < 4; ++n) {
    const int d = n * 16 + lh;
#pragma unroll
    for (int j = 0; j < 8; ++j) {
      if (ebase + j + 8 * hi < n_edges) {
        float v = acc[n][j] * x[(long long)s_idx[j] * DIM + d];
        atomic_add_f32(&msg[(long long)t_idx[j] * DIM + d], v);
      }
    }
  }

  // Degree: one lane per edge of the tile.
  if (hi == 0 && (ebase + lh) < n_edges) {
    int t = ei[(long long)n_edges + ebase + lh];
    atomic_add_f32(&deg[t], 1.0f);
  }
}

// ---------------------------------------------------------------------------
// Node kernel: out = msg/max(deg,1) + x@root + bias. One wave per 16 nodes.
// ---------------------------------------------------------------------------
__global__ __launch_bounds__(256) void rgcn_node_kernel(
    const float* __restrict__ x,      // [N,64]
    const float* __restrict__ root,   // [64,64]
    const float* __restrict__ bias,   // [64]
    const float* __restrict__ msg,    // [N,64]
    const float* __restrict__ deg,    // [N]
    float* __restrict__ out,          // [N,64]
    int n_nodes)
{
  __shared__ float srtT[DIM * LDS_PITCH];
  for (int idx = threadIdx.x; idx < DIM * DIM; idx += 256) {
    int r = idx >> 6, c = idx & 63;
    srtT[c * LDS_PITCH + r] = root[idx];
  }
  __syncthreads();

  const int lane = threadIdx.x & 31;
  const int lh   = lane & 15;
  const int hi   = lane >> 4;
  const long long tile  = (long long)blockIdx.x * 8 + (threadIdx.x >> 5);
  const long long nbase = tile * 16;
  if (nbase >= n_nodes) return;                 // wave-uniform

  long long arow_n = nbase + lh;
  if (arow_n >= n_nodes) arow_n = n_nodes - 1;
  const float* arow = x + arow_n * DIM + 2 * hi;

  v8f acc[4] = {v8f{}, v8f{}, v8f{}, v8f{}};

#pragma unroll
  for (int k = 0; k < DIM; k += 4) {
    v2f a;
    a.x = arow[k];
    a.y = arow[k + 1];
    const int kk = k + 2 * hi;
#pragma unroll
    for (int n = 0; n < 4; ++n) {
      v2f b;
      b.x = srtT[(n * 16 + lh) * LDS_PITCH + kk];
      b.y = srtT[(n * 16 + lh) * LDS_PITCH + kk + 1];
      acc[n] = wmma_f32_16x16x4(a, b, acc[n]);
    }
  }

#pragma unroll
  for (int n = 0; n < 4; ++n) {
    const int d = n * 16 + lh;
    const float bv = bias[d];
#pragma unroll
    for (int j = 0; j < 8; ++j) {
      long long row = nbase + j + 8 * hi;
      if (row < n_nodes) {
        float dg  = deg[row];
        float agg = msg[row * DIM + d] / fmaxf(dg, 1.0f);
        out[row * DIM + d] = agg + acc[n][j] + bv;
      }
    }
  }
}

// ---------------------------------------------------------------------------
extern "C" void kernel_launch(void* const* d_in, const int* in_sizes, int n_in,
                              void* d_out, int out_size, void* d_ws, size_t ws_size,
                              hipStream_t stream) {
  const float* x    = (const float*)d_in[0];
  const int*   ei   = (const int*)  d_in[1];   // [2,E]
  const float* emb  = (const float*)d_in[2];
  const float* lw   = (const float*)d_in[3];
  const float* root = (const float*)d_in[4];
  const float* bias = (const float*)d_in[5];

  const int n_nodes = in_sizes[0] / DIM;
  const int n_edges = in_sizes[2] / DIM;

  float* msg = (float*)d_ws;                       // [N,64]
  float* deg = msg + (long long)n_nodes * DIM;     // [N]

  // Zero scatter buffers (harness does not re-zero between replays).
  long long nz = (long long)n_nodes * (DIM + 1);
  zero_ws_kernel<<<(int)((nz + 255) / 256), 256, 0, stream>>>(msg, nz);

  const long long etiles = ((long long)n_edges + 15) / 16;
  rgcn_edge_kernel<<<(int)((etiles + 7) / 8), 256, 0, stream>>>(
      x, ei, emb, lw, msg, deg, n_edges);

  const long long ntiles = ((long long)n_nodes + 15) / 16;
  rgcn_node_kernel<<<(int)((ntiles + 7) / 8), 256, 0, stream>>>(
      x, root, bias, msg, deg, (float*)d_out, n_nodes);
}